// AutoCorrelation_61710090109793
// MI455X (gfx1250) — compile-verified
//
#include <hip/hip_runtime.h>
#include <math.h>

// Problem constants (B, L, V, H, E) = (4, 96, 207, 8, 64), top_k = int(log(96)) = 4
#define B_DIM 4
#define L_DIM 96
#define V_DIM 207
#define H_DIM 8
#define E_DIM 64
#define C_DIM 512                 // H*E channel dim
#define TSTR  (V_DIM * C_DIM)     // element stride between consecutive time steps: 105984
#define CHUNK 64                  // K-chunk staged in LDS
#define NCH   (C_DIM / CHUNK)     // 8 chunks
#define SP    68                  // padded LDS panel row stride (floats): 96*68*4*2 = 52KB
#define SPS   97                  // padded LDS stride for the 96x96 score matrix
#define TOPK  4

typedef __attribute__((ext_vector_type(2))) float v2f;
typedef __attribute__((ext_vector_type(8))) float v8f;

#if __has_builtin(__builtin_amdgcn_global_load_async_to_lds_b128)
#define HAVE_ASYNC_LDS 1
// Parameter types per hipcc diagnostic: arg0 = global (__device__, AS1) int4*,
// arg1 = LDS (__shared__, AS3) int4*, then two immediate ints (offset, cpol).
typedef __attribute__((__vector_size__(16))) int i4vec;
typedef __attribute__((address_space(1))) i4vec* gas_ptr;
typedef __attribute__((address_space(3))) i4vec* las_ptr;
#else
#define HAVE_ASYNC_LDS 0
#endif

__device__ __forceinline__ void wait_async_all() {
#if __has_builtin(__builtin_amdgcn_s_wait_asynccnt)
  __builtin_amdgcn_s_wait_asynccnt(0);
#else
  asm volatile("s_wait_asynccnt 0x0" ::: "memory");
#endif
}

// ---------------------------------------------------------------------------
// Stage 1: per-(b,v) Gram matrix S = Q * K^T (96x96, K=512) via f32 WMMA,
// then circular-diagonal reduction -> mean_value[b,v,tau] (exactly matches
// mean over H,E of irfft(rfft(q)*conj(rfft(k))) up to fp rounding).
// 192 threads = 6 waves; wave w owns M-panel rows [16w, 16w+16).
// ---------------------------------------------------------------------------
__global__ __launch_bounds__(192) void corr_gram_kernel(
    const float* __restrict__ q, const float* __restrict__ k,
    float* __restrict__ mv)
{
  __shared__ float smem[2 * L_DIM * SP];   // Q/K panels; reused for S at the end
  float* lq = smem;
  float* lk = smem + L_DIM * SP;

  const int tid  = threadIdx.x;
  const int bv   = blockIdx.x;
  const int b    = bv / V_DIM;
  const int v    = bv % V_DIM;
  const size_t base = ((size_t)b * L_DIM * V_DIM + v) * C_DIM;

  const int wv   = tid >> 5;               // wave 0..5 (wave32)
  const int lane = tid & 31;
  const int ln   = lane & 15;
  const int ko   = (lane >> 4) << 1;       // K sub-offset: 0 (lanes 0-15) / 2 (16-31)
  const int m0   = wv * 16;                // this wave's M-panel

  v8f acc[6];
  const v8f vzero = {0.f, 0.f, 0.f, 0.f, 0.f, 0.f, 0.f, 0.f};
#pragma unroll
  for (int n = 0; n < 6; ++n) acc[n] = vzero;

  for (int ch = 0; ch < NCH; ++ch) {
    const int c0 = ch * CHUNK;
    __syncthreads();
    // Stage 96 rows x 64 floats per panel into LDS: 1536 16B moves, 8/thread.
#pragma unroll
    for (int i = 0; i < 8; ++i) {
      const int t   = tid + i * 192;
      const int row = t >> 4;
      const int c4  = (t & 15) << 2;
      const float* gq = q + base + (size_t)row * TSTR + c0 + c4;
      const float* gk = k + base + (size_t)row * TSTR + c0 + c4;
      float* dq = &lq[row * SP + c4];
      float* dk = &lk[row * SP + c4];
#if HAVE_ASYNC_LDS
      // Direct cache->LDS DMA, tracked by ASYNCcnt (no VGPR staging).
      __builtin_amdgcn_global_load_async_to_lds_b128((gas_ptr)gq, (las_ptr)dq, 0, 0);
      __builtin_amdgcn_global_load_async_to_lds_b128((gas_ptr)gk, (las_ptr)dk, 0, 0);
#else
      *(float4*)dq = *(const float4*)gq;
      *(float4*)dk = *(const float4*)gk;
#endif
    }
#if HAVE_ASYNC_LDS
    wait_async_all();
#endif
    __syncthreads();
    // 16 K-steps of 4. Batch A + all 6 B fragments into registers first so the
    // six WMMAs issue back-to-back behind a single DScnt wait.
#pragma unroll
    for (int ks = 0; ks < CHUNK / 4; ++ks) {
      const int cc = ks * 4 + ko;
      const v2f a = *(const v2f*)&lq[(m0 + ln) * SP + cc];
      v2f bb[6];
#pragma unroll
      for (int n = 0; n < 6; ++n)
        bb[n] = *(const v2f*)&lk[(n * 16 + ln) * SP + cc];
#pragma unroll
      for (int n = 0; n < 6; ++n)
        acc[n] = __builtin_amdgcn_wmma_f32_16x16x4_f32(
            /*neg_a=*/false, a, /*neg_b=*/false, bb[n],
            /*c_mod=*/(short)0, acc[n], /*reuse_a=*/false, /*reuse_b=*/false);
    }
  }

  // Spill S tiles to LDS (deterministic, no atomics), then reduce diagonals.
  __syncthreads();
  float* S = smem;                          // 96*97 = 9312 floats <= 2*96*68
#pragma unroll
  for (int n = 0; n < 6; ++n) {
    const int nn = n * 16 + ln;
#pragma unroll
    for (int r = 0; r < 8; ++r) {
      const int m = m0 + r + ((lane >> 4) << 3);   // C layout: M = r + 8*(lane/16)
      S[m * SPS + nn] = acc[n][r];
    }
  }
  __syncthreads();
  if (tid < L_DIM) {
    float s = 0.f;
    for (int t = 0; t < L_DIM; ++t) {
      int row = t + tid;
      if (row >= L_DIM) row -= L_DIM;       // corr[tau] = sum_t S[(t+tau)%L, t]
      s += S[row * SPS + t];
    }
    mv[(size_t)bv * L_DIM + tid] = s * (1.0f / (float)C_DIM);
  }
}

// ---------------------------------------------------------------------------
// Stage 2: per-v top-4 over sum_b mean_value (order == mean over b), then
// per-(b,v) softmax over the gathered weights.
// ---------------------------------------------------------------------------
__global__ __launch_bounds__(96) void topk_softmax_kernel(
    const float* __restrict__ mv, int* __restrict__ idxw, float* __restrict__ tw)
{
  const int v   = blockIdx.x;
  const int tid = threadIdx.x;
  __shared__ float sg[L_DIM];
  __shared__ int   sidx[TOPK];

  float s = 0.f;
#pragma unroll
  for (int b = 0; b < B_DIM; ++b)
    s += mv[((size_t)(b * V_DIM + v)) * L_DIM + tid];
  sg[tid] = s;
  __syncthreads();

  if (tid == 0) {
    for (int kk = 0; kk < TOPK; ++kk) {
      float best = -INFINITY; int bi = 0;
      for (int i = 0; i < L_DIM; ++i) {
        if (sg[i] > best) { best = sg[i]; bi = i; }   // strict > == lowest-index tie-break
      }
      sidx[kk] = bi;
      sg[bi] = -INFINITY;
    }
  }
  __syncthreads();

  if (tid < TOPK) idxw[v * TOPK + tid] = sidx[tid];
  if (tid < B_DIM) {
    const int b = tid;
    float w[TOPK];
#pragma unroll
    for (int kk = 0; kk < TOPK; ++kk)
      w[kk] = mv[((size_t)(b * V_DIM + v)) * L_DIM + sidx[kk]];
    float m = fmaxf(fmaxf(w[0], w[1]), fmaxf(w[2], w[3]));
    float e0 = expf(w[0] - m), e1 = expf(w[1] - m);
    float e2 = expf(w[2] - m), e3 = expf(w[3] - m);
    const float inv = 1.0f / (e0 + e1 + e2 + e3);
    float* o = tw + ((size_t)(b * V_DIM + v)) * TOPK;
    o[0] = e0 * inv; o[1] = e1 * inv; o[2] = e2 * inv; o[3] = e3 * inv;
  }
}

// ---------------------------------------------------------------------------
// Stage 3: out[b,l,v,e,h] = sum_k tmp[b,v,k] * values[b,(l+idx[v,k])%L, v, h, e]
// One block per (b,v): output-coalesced stores; gathered reads stay in L0/L2
// (each 2KB value row is fully consumed by the block).
// ---------------------------------------------------------------------------
__global__ __launch_bounds__(256) void agg_kernel(
    const float* __restrict__ vals, const int* __restrict__ idxw,
    const float* __restrict__ tw, float* __restrict__ out)
{
  const int bv = blockIdx.x;
  const int b  = bv / V_DIM;
  const int v  = bv % V_DIM;
  const int tid = threadIdx.x;

  float w[TOPK]; int d[TOPK];
#pragma unroll
  for (int kk = 0; kk < TOPK; ++kk) {
    w[kk] = tw[((size_t)(b * V_DIM + v)) * TOPK + kk];
    d[kk] = idxw[v * TOPK + kk];
  }
  const size_t base = ((size_t)b * L_DIM * V_DIM + v) * C_DIM;

  for (int l = 0; l < L_DIM; ++l) {
    const float* r[TOPK];
#pragma unroll
    for (int kk = 0; kk < TOPK; ++kk) {
      int t = l + d[kk];
      if (t >= L_DIM) t -= L_DIM;
      r[kk] = vals + base + (size_t)t * TSTR;
    }
    float* orow = out + base + (size_t)l * TSTR;
#pragma unroll
    for (int j = 0; j < 2; ++j) {
      const int o   = tid + j * 256;       // output channel e*8+h
      const int e   = o >> 3;
      const int h   = o & 7;
      const int cin = h * E_DIM + e;       // input channel h*64+e
      float s = w[0] * r[0][cin];
      s = fmaf(w[1], r[1][cin], s);
      s = fmaf(w[2], r[2][cin], s);
      s = fmaf(w[3], r[3][cin], s);
      orow[o] = s;
    }
  }
}

// ---------------------------------------------------------------------------
extern "C" void kernel_launch(void* const* d_in, const int* in_sizes, int n_in,
                              void* d_out, int out_size, void* d_ws, size_t ws_size,
                              hipStream_t stream) {
  const float* q    = (const float*)d_in[0];
  const float* kk   = (const float*)d_in[1];
  const float* vals = (const float*)d_in[2];
  float* out = (float*)d_out;

  // Workspace layout (all 16B aligned):
  //   mv  : float[B*V*96]  = 317,952 B
  //   idx : int  [V*4]     =   3,312 B
  //   tw  : float[B*V*4]   =  13,248 B
  char* ws = (char*)d_ws;
  float* mv   = (float*)ws;
  int*   idxw = (int*)(ws + (size_t)B_DIM * V_DIM * L_DIM * sizeof(float));
  float* tw   = (float*)(ws + (size_t)B_DIM * V_DIM * L_DIM * sizeof(float)
                            + (size_t)V_DIM * TOPK * sizeof(int));

  corr_gram_kernel<<<B_DIM * V_DIM, 192, 0, stream>>>(q, kk, mv);
  topk_softmax_kernel<<<V_DIM, 96, 0, stream>>>(mv, idxw, tw);
  agg_kernel<<<B_DIM * V_DIM, 256, 0, stream>>>(vals, idxw, tw, out);
}